// GNNModel_90134183674653
// MI455X (gfx1250) — compile-verified
//
#include <hip/hip_runtime.h>
#include <hip/hip_bf16.h>

// ---------------------------------------------------------------------------
// GNN with gated aggregation (sum/mean/max), per-layer FFN (two 128x128 GEMMs
// via V_WMMA_F32_16X16X4_F32), skip connections, and gated global pooling.
// ---------------------------------------------------------------------------

typedef float v8f __attribute__((ext_vector_type(8)));
typedef float v2f __attribute__((ext_vector_type(2)));

#define GNN_D 128
#define TILE_M 16
#define LDS_STRIDE 132   // 132 mod 64 = 4 -> column reads hit 16 distinct banks

// order-preserving float <-> uint map (for atomicMax on floats)
__device__ __forceinline__ unsigned f2mono(float f) {
    unsigned u = __float_as_uint(f);
    return (u & 0x80000000u) ? ~u : (u | 0x80000000u);
}
__device__ __forceinline__ float mono2f(unsigned m) {
    unsigned u = (m & 0x80000000u) ? (m & 0x7FFFFFFFu) : ~m;
    return __uint_as_float(u);
}
#define MONO_NEG_INF 0x007FFFFFu   // f2mono(-inf)

// -------------------------- small utility kernels --------------------------

__global__ __launch_bounds__(256) void gnn_init_mono_kernel(unsigned* __restrict__ p, int n) {
    int i = blockIdx.x * 256 + threadIdx.x;
    if (i < n) p[i] = MONO_NEG_INF;
}

__global__ __launch_bounds__(256) void gnn_deg_kernel(const int* __restrict__ dst,
                                                      float* __restrict__ deg, int E) {
    int e = blockIdx.x * 256 + threadIdx.x;
    if (e < E) atomicAdd(&deg[dst[e]], 1.0f);
}

// ------------------------- edge scatter (aggregate) -------------------------
// one wave32 per edge; each lane handles 4 of the 128 features.
__global__ __launch_bounds__(256) void gnn_scatter_kernel(
    const float* __restrict__ x, const int* __restrict__ src,
    const int* __restrict__ dst, float* __restrict__ s,
    unsigned* __restrict__ mx, int E)
{
    int gid  = blockIdx.x * 256 + threadIdx.x;
    int e    = gid >> 5;
    int lane = gid & 31;
    if (e >= E) return;
    int sn = src[e];
    int dn = dst[e];
    const float4 v = *reinterpret_cast<const float4*>(x + (size_t)sn * GNN_D + lane * 4);
    float*    sp = s  + (size_t)dn * GNN_D + lane * 4;
    unsigned* mp = mx + (size_t)dn * GNN_D + lane * 4;
    atomicAdd(sp + 0, v.x); atomicAdd(sp + 1, v.y);
    atomicAdd(sp + 2, v.z); atomicAdd(sp + 3, v.w);
    atomicMax(mp + 0, f2mono(v.x)); atomicMax(mp + 1, f2mono(v.y));
    atomicMax(mp + 2, f2mono(v.z)); atomicMax(mp + 3, f2mono(v.w));
}

// --------------------- fused combine + FFN layer kernel ---------------------
// block = 256 threads = 8 waves, one 16-row tile of x per block.
// wave w computes output columns [16w, 16w+16) with 16x16x4 fp32 WMMA.
__global__ __launch_bounds__(256) void gnn_layer_kernel(
    const float* __restrict__ x, const float* __restrict__ s,
    const unsigned* __restrict__ mx, const float* __restrict__ deg,
    const float* __restrict__ zagg,
    const float* __restrict__ W1, const float* __restrict__ b1,
    const float* __restrict__ W2, const float* __restrict__ b2,
    float* __restrict__ xout)
{
    __shared__ float hs[TILE_M * LDS_STRIDE];   // h_in tile (A of GEMM1)
    __shared__ float xt[TILE_M * LDS_STRIDE];   // x tile (for skip)
    __shared__ float ts[TILE_M * LDS_STRIDE];   // relu(h@W1+b1) (A of GEMM2)

    const int tid  = threadIdx.x;
    const int row0 = blockIdx.x * TILE_M;

    const float z0 = zagg[0], z1 = zagg[1], z2 = zagg[2];

    // Phase 1: build h_in = x + z0*s + z1*s/deg_c + z2*(deg>0 ? mx : 0)
    {
        int base = tid * 8;              // 2048 elements / 256 threads
        int r = base >> 7;
        int c = base & 127;
        size_t g = (size_t)(row0 + r) * GNN_D + c;
        float dg = deg[row0 + r];
        float dc = fmaxf(dg, 1.0f);
        bool  nb = dg > 0.0f;
        #pragma unroll
        for (int q = 0; q < 2; ++q) {
            float4 xv = *reinterpret_cast<const float4*>(x + g + q * 4);
            float4 sv = *reinterpret_cast<const float4*>(s + g + q * 4);
            uint4  mv = *reinterpret_cast<const uint4*>(mx + g + q * 4);
            float4 h;
            h.x = xv.x + z0 * sv.x + z1 * (sv.x / dc) + z2 * (nb ? mono2f(mv.x) : 0.0f);
            h.y = xv.y + z0 * sv.y + z1 * (sv.y / dc) + z2 * (nb ? mono2f(mv.y) : 0.0f);
            h.z = xv.z + z0 * sv.z + z1 * (sv.z / dc) + z2 * (nb ? mono2f(mv.z) : 0.0f);
            h.w = xv.w + z0 * sv.w + z1 * (sv.w / dc) + z2 * (nb ? mono2f(mv.w) : 0.0f);
            *reinterpret_cast<float4*>(&hs[r * LDS_STRIDE + c + q * 4]) = h;
            *reinterpret_cast<float4*>(&xt[r * LDS_STRIDE + c + q * 4]) = xv;
        }
    }
    __syncthreads();

    const int wave  = tid >> 5;
    const int lane  = tid & 31;
    const int ncol  = (wave << 4) + (lane & 15);     // output column 0..127
    const int mlo   = lane & 15;                     // A row for this lane
    const int koff  = (lane >> 4) << 1;              // lanes 16-31 carry K+2,K+3
    const int rbase = (lane >> 4) << 3;              // C/D rows: 0..7 or 8..15
    const float bia1 = b1[ncol];
    const float bia2 = b2[ncol];

    // GEMM1: t = relu(h_in @ W1 + b1)
    v8f acc = {};
    #pragma unroll
    for (int k0 = 0; k0 < GNN_D; k0 += 4) {
        int kk = k0 + koff;
        v2f a; a.x = hs[mlo * LDS_STRIDE + kk];
               a.y = hs[mlo * LDS_STRIDE + kk + 1];
        v2f b; b.x = W1[(size_t)kk * GNN_D + ncol];
               b.y = W1[(size_t)(kk + 1) * GNN_D + ncol];
        acc = __builtin_amdgcn_wmma_f32_16x16x4_f32(false, a, false, b,
                                                    (short)0, acc, false, false);
    }
    #pragma unroll
    for (int i = 0; i < 8; ++i)
        ts[(rbase + i) * LDS_STRIDE + ncol] = fmaxf(acc[i] + bia1, 0.0f);
    __syncthreads();

    // GEMM2: h2 = t @ W2 + b2 ; x_out = x + relu(h2)
    v8f acc2 = {};
    #pragma unroll
    for (int k0 = 0; k0 < GNN_D; k0 += 4) {
        int kk = k0 + koff;
        v2f a; a.x = ts[mlo * LDS_STRIDE + kk];
               a.y = ts[mlo * LDS_STRIDE + kk + 1];
        v2f b; b.x = W2[(size_t)kk * GNN_D + ncol];
               b.y = W2[(size_t)(kk + 1) * GNN_D + ncol];
        acc2 = __builtin_amdgcn_wmma_f32_16x16x4_f32(false, a, false, b,
                                                     (short)0, acc2, false, false);
    }
    #pragma unroll
    for (int i = 0; i < 8; ++i) {
        int r = rbase + i;
        float h2 = acc2[i] + bia2;
        float xv = xt[r * LDS_STRIDE + ncol];
        xout[(size_t)(row0 + r) * GNN_D + ncol] = xv + fmaxf(h2, 0.0f);
    }
}

// ------------------------------ global pooling ------------------------------
__global__ __launch_bounds__(256) void gnn_pool_scatter_kernel(
    const float* __restrict__ x, const int* __restrict__ batch,
    float* __restrict__ ps, unsigned* __restrict__ pmx,
    float* __restrict__ cnt, int N)
{
    int gid  = blockIdx.x * 256 + threadIdx.x;
    int n    = gid >> 5;
    int lane = gid & 31;
    if (n >= N) return;
    int b = batch[n];
    const float4 v = *reinterpret_cast<const float4*>(x + (size_t)n * GNN_D + lane * 4);
    float*    sp = ps  + (size_t)b * GNN_D + lane * 4;
    unsigned* mp = pmx + (size_t)b * GNN_D + lane * 4;
    atomicAdd(sp + 0, v.x); atomicAdd(sp + 1, v.y);
    atomicAdd(sp + 2, v.z); atomicAdd(sp + 3, v.w);
    atomicMax(mp + 0, f2mono(v.x)); atomicMax(mp + 1, f2mono(v.y));
    atomicMax(mp + 2, f2mono(v.z)); atomicMax(mp + 3, f2mono(v.w));
    if (lane == 0) atomicAdd(&cnt[b], 1.0f);
}

__global__ __launch_bounds__(256) void gnn_pool_finalize_kernel(
    const float* __restrict__ ps, const unsigned* __restrict__ pmx,
    const float* __restrict__ cnt, const float* __restrict__ zpool,
    float* __restrict__ out, int total)
{
    int i = blockIdx.x * 256 + threadIdx.x;
    if (i >= total) return;
    int b = i >> 7;  // / 128
    float c  = cnt[b];
    float sv = ps[i];
    float mv = (c > 0.0f) ? mono2f(pmx[i]) : 0.0f;
    out[i] = zpool[0] * sv + zpool[1] * (sv / fmaxf(c, 1.0f)) + zpool[2] * mv;
}

// --------------------------------- launcher ---------------------------------

extern "C" void kernel_launch(void* const* d_in, const int* in_sizes, int n_in,
                              void* d_out, int out_size, void* d_ws, size_t ws_size,
                              hipStream_t stream) {
    const float* x_in   = (const float*)d_in[0];
    const int*   eidx   = (const int*)d_in[1];    // [2][E]
    const int*   batch  = (const int*)d_in[2];    // [N]
    const float* z_agg  = (const float*)d_in[3];  // [L][3]
    const float* z_pool = (const float*)d_in[4];  // [3]
    const float* W1     = (const float*)d_in[5];  // [L][D][D]
    const float* b1     = (const float*)d_in[6];  // [L][D]
    const float* W2     = (const float*)d_in[7];  // [L][D][D]
    const float* b2     = (const float*)d_in[8];  // [L][D]
    (void)n_in;

    const int N = in_sizes[0] / GNN_D;
    const int E = in_sizes[1] / 2;
    const int L = in_sizes[5] / (GNN_D * GNN_D);
    const int Bg = out_size / GNN_D;
    const size_t ND = (size_t)N * GNN_D;
    const size_t BD = (size_t)Bg * GNN_D;

    // carve workspace (256B-aligned chunks)
    char* p = (char*)d_ws;
    char* pend = p + ws_size;
    auto take = [&](size_t bytes) -> void* {
        void* r = (void*)p;
        p += (bytes + 255) & ~(size_t)255;
        return r;
    };
    float*    xa   = (float*)take(ND * 4);
    float*    xb   = (float*)take(ND * 4);
    float*    s    = (float*)take(ND * 4);
    unsigned* mx   = (unsigned*)take(ND * 4);
    float*    deg  = (float*)take((size_t)N * 4);
    float*    ps   = (float*)take(BD * 4);
    unsigned* pmx  = (unsigned*)take(BD * 4);
    float*    cnt  = (float*)take((size_t)Bg * 4);
    if (p > pend) return;  // workspace too small; bail safely

    const int* src = eidx;
    const int* dst = eidx + E;

    hipMemcpyAsync(xa, x_in, ND * 4, hipMemcpyDeviceToDevice, stream);

    // degrees (same every layer)
    hipMemsetAsync(deg, 0, (size_t)N * 4, stream);
    gnn_deg_kernel<<<(E + 255) / 256, 256, 0, stream>>>(dst, deg, E);

    float* xcur = xa;
    float* xnxt = xb;
    for (int l = 0; l < L; ++l) {
        hipMemsetAsync(s, 0, ND * 4, stream);
        gnn_init_mono_kernel<<<(int)((ND + 255) / 256), 256, 0, stream>>>(mx, (int)ND);
        gnn_scatter_kernel<<<(E + 7) / 8, 256, 0, stream>>>(xcur, src, dst, s, mx, E);
        gnn_layer_kernel<<<N / TILE_M, 256, 0, stream>>>(
            xcur, s, mx, deg, z_agg + l * 3,
            W1 + (size_t)l * GNN_D * GNN_D, b1 + (size_t)l * GNN_D,
            W2 + (size_t)l * GNN_D * GNN_D, b2 + (size_t)l * GNN_D, xnxt);
        float* t = xcur; xcur = xnxt; xnxt = t;
    }

    // global pooling
    hipMemsetAsync(ps, 0, BD * 4, stream);
    hipMemsetAsync(cnt, 0, (size_t)Bg * 4, stream);
    gnn_init_mono_kernel<<<(int)((BD + 255) / 256), 256, 0, stream>>>(pmx, (int)BD);
    gnn_pool_scatter_kernel<<<(N + 7) / 8, 256, 0, stream>>>(xcur, batch, ps, pmx, cnt, N);
    gnn_pool_finalize_kernel<<<(int)((BD + 255) / 256), 256, 0, stream>>>(
        ps, pmx, cnt, z_pool, (float*)d_out, (int)BD);
}